// TanhConvolvedAttnLayer_50173807952856
// MI455X (gfx1250) — compile-verified
//
#include <hip/hip_runtime.h>
#include <cmath>

typedef __attribute__((ext_vector_type(2))) float v2f;
typedef __attribute__((ext_vector_type(4))) float v4f;
typedef __attribute__((ext_vector_type(8))) float v8f;

namespace {
constexpr int kB = 2, kS = 4096, kH = 8, kO = 64, kW = 128, kP = 64, kC = 32;
constexpr int kHD = 512;              // H*D floats per (b,s) row of q/k/v
constexpr int LDW = 68;               // padded stride for 64-wide LDS buffers
constexpr int SLAB_LD = 148;          // padded stride for 144-wide attn slab

// LDS layout (floats)
constexpr int OFF_WQ = 0;                       // 64 x LDW
constexpr int OFF_WK = OFF_WQ + 64 * LDW;       // 64 x LDW
constexpr int OFF_BQ = OFF_WK + 64 * LDW;       // 64
constexpr int OFF_BK = OFF_BQ + 64;             // 64
constexpr int OFF_OC = OFF_BK + 64;             // 129*8
constexpr int OFF_KB = OFF_OC + 129 * 8;        // 256 x LDW  projected k'
constexpr int OFF_QB = OFF_KB + 256 * LDW;      // 128 x LDW  projected q'
constexpr int OFF_VB = OFF_QB + 128 * LDW;      // 256 x LDW  raw-k scratch, later v
constexpr int OFF_SL = OFF_VB + 256 * LDW;      // slabs (8 x 16 x SLAB_LD); raw-q staging early
constexpr int SMEM_FLOATS = OFF_SL + 8 * 16 * SLAB_LD;   // 72328 floats = 289312 B
}

// ---- CDNA5 async global->LDS copy (ASYNCcnt), EXEC-masked per lane ----
__device__ __forceinline__ void async_copy_b128(void* lds_dst, const void* gsrc) {
  // Low 32 bits of a generic LDS pointer are the LDS byte address (ISA 10.2).
  unsigned int l = (unsigned int)(unsigned long long)lds_dst;
  asm volatile("global_load_async_to_lds_b128 %0, %1, off"
               :: "v"(l), "v"(gsrc)
               : "memory");
}
__device__ __forceinline__ void wait_async0() {
  asm volatile("s_wait_asynccnt 0x0" ::: "memory");
}

// A-style fragment: element [row0 + lane%16][k0 + 2*(lane/16) + p] from row-major storage.
// Also serves as B when B[k][n] = Mat[n][k] (scores q*k^T, W^T in projections).
__device__ __forceinline__ v2f frag_rows(const float* base, int row0, int k0, int ld, int lane) {
  const float* p = base + (size_t)(row0 + (lane & 15)) * ld + k0 + ((lane >> 4) << 1);
  return *(const v2f*)p;  // 8B aligned: ld even, k0 % 4 == 0
}

// B fragment from row-major [K][N] storage (attn @ v)
__device__ __forceinline__ v2f frag_cols(const float* base, int k0, int n0, int ld, int lane) {
  int r = k0 + ((lane >> 4) << 1);
  int n = n0 + (lane & 15);
  v2f f;
  f.x = base[(size_t)r * ld + n];
  f.y = base[(size_t)(r + 1) * ld + n];
  return f;
}

__device__ __forceinline__ v8f wmma4(v2f a, v2f b, v8f c) {
  // fp32 matrix core: D(16x16) = A(16x4) * B(4x16) + C
  return __builtin_amdgcn_wmma_f32_16x16x4_f32(false, a, false, b, (short)0, c, false, false);
}

// One 16x16 projection tile: dst[m0+m][n0+n] = raw[m0+m][:] . Wl[n0+n][:] + bl[n0+n]
__device__ __forceinline__ void project_tile(const float* raw, const float* Wl, const float* bl,
                                             float* dst, int m0, int n0, int lane) {
  float bv = bl[n0 + (lane & 15)];
  v8f acc;
#pragma unroll
  for (int r = 0; r < 8; ++r) acc[r] = bv;
#pragma unroll
  for (int k0 = 0; k0 < 64; k0 += 4) {
    v2f a = frag_rows(raw, m0, k0, LDW, lane);
    v2f b = frag_rows(Wl, n0, k0, LDW, lane);   // B[k][n] = W[n][k]
    acc = wmma4(a, b, acc);
  }
  int n = lane & 15, hi = lane >> 4;
#pragma unroll
  for (int r = 0; r < 8; ++r)
    dst[(size_t)(m0 + r + 8 * hi) * LDW + n0 + n] = acc[r];
}

__global__ void __launch_bounds__(256)
zero_kernel(float* p, int n) {
  int i = blockIdx.x * blockDim.x + threadIdx.x;
  if (i < n) p[i] = 0.0f;
}

__global__ void __launch_bounds__(256)
banded_attn_kernel(const float* __restrict__ qg, const float* __restrict__ kg,
                   const float* __restrict__ vg, const float* __restrict__ Wq,
                   const float* __restrict__ bq, const float* __restrict__ Wk,
                   const float* __restrict__ bk, const float* __restrict__ ocpe,
                   float* __restrict__ attn_out, float* __restrict__ out) {
  extern __shared__ float smem[];
  float* WqL = smem + OFF_WQ;
  float* WkL = smem + OFF_WK;
  float* bqL = smem + OFF_BQ;
  float* bkL = smem + OFF_BK;
  float* ocL = smem + OFF_OC;
  float* kbuf = smem + OFF_KB;
  float* qbuf = smem + OFF_QB;
  float* vbuf = smem + OFF_VB;   // raw-k scratch, then v window
  float* slab = smem + OFF_SL;   // raw-q staging in phase 0/1, attn slabs later

  const int tid = threadIdx.x;
  const int lane = tid & 31;
  const int wave = tid >> 5;
  const int bid = blockIdx.x;          // (b*C + c)*H + h
  const int h = bid & 7;
  const int c = (bid >> 3) & 31;
  const int b = bid >> 8;
  const int base_j = c * kW - kP;      // global pos of window row 0

  // ================= phase 0: kick off async staging + load constants ========
  // raw k window (256 x 64, zero-padded halo) -> vbuf  [async]
  for (int e = tid; e < 256 * 16; e += 256) {
    int row = e >> 4, c4 = (e & 15) * 4;
    int j = base_j + row;
    float* dst = &vbuf[row * LDW + c4];
    if (j >= 0 && j < kS)
      async_copy_b128(dst, &kg[(size_t)(b * kS + j) * kHD + h * 64 + c4]);
    else
      *(v4f*)dst = (v4f){0.f, 0.f, 0.f, 0.f};
  }
  // raw q center rows (128 x 64, always valid) -> slab region  [async]
  for (int e = tid; e < 128 * 16; e += 256) {
    int row = e >> 4, c4 = (e & 15) * 4;
    int i = c * kW + row;
    async_copy_b128(&slab[row * LDW + c4],
                    &qg[(size_t)(b * kS + i) * kHD + h * 64 + c4]);
  }
  // constants: Wq, Wk (row-major, padded), biases, ocpe
  for (int e = tid; e < 64 * 16; e += 256) {
    int r = e >> 4, c4 = (e & 15) * 4;
    *(v4f*)&WqL[r * LDW + c4] = *(const v4f*)&Wq[r * 64 + c4];
    *(v4f*)&WkL[r * LDW + c4] = *(const v4f*)&Wk[r * 64 + c4];
  }
  if (tid < 64) { bqL[tid] = bq[tid]; bkL[tid] = bk[tid]; }
  for (int e = tid; e < 129 * 8; e += 256) ocL[e] = ocpe[e];

  wait_async0();
  __syncthreads();

  // ================= phase 1: fused projections (96 WMMA tiles / 8 waves) ====
  // tiles 0..63:  k' (16 row-tiles x 4 col-tiles) ; tiles 64..95: q' (8 x 4)
  for (int t = 0; t < 12; ++t) {
    int tIdx = wave * 12 + t;
    if (tIdx < 64)
      project_tile(vbuf, WkL, bkL, kbuf, (tIdx >> 2) * 16, (tIdx & 3) * 16, lane);
    else {
      int u = tIdx - 64;
      project_tile(slab, WqL, bqL, qbuf, (u >> 2) * 16, (u & 3) * 16, lane);
    }
  }
  __syncthreads();

  // ================= phase 2: async v load overlapped with scores ============
  // v window (256 x 64, zero-padded) -> vbuf [async]; not needed until phase 3
  for (int e = tid; e < 256 * 16; e += 256) {
    int row = e >> 4, c4 = (e & 15) * 4;
    int j = base_j + row;
    float* dst = &vbuf[row * LDW + c4];
    if (j >= 0 && j < kS)
      async_copy_b128(dst, &vg[(size_t)(b * kS + j) * kHD + h * 64 + c4]);
    else
      *(v4f*)dst = (v4f){0.f, 0.f, 0.f, 0.f};
  }

  // scores (banded) + bias + mask + tanh; stream band to global, stage slab
  float* myslab = slab + wave * (16 * SLAB_LD);
  const int x0w = 16 * wave;  // qbuf row base; window x = 64 + x0w + m
  const int nl = lane & 15, hi = lane >> 4;

  // hoist the wave's q' A-fragments (reused by all 9 column tiles)
  v2f qf[16];
#pragma unroll
  for (int kk = 0; kk < 16; ++kk)
    qf[kk] = frag_rows(qbuf, x0w, kk * 4, LDW, lane);

  for (int ty = 0; ty < 9; ++ty) {           // band: cols y in [16w, 16w+144)
    const int ycol0 = x0w + 16 * ty;         // kbuf row base for this tile
    v8f acc;
#pragma unroll
    for (int r = 0; r < 8; ++r) {
      int m = r + 8 * hi;
      int t = 16 * ty + nl - m;              // = (j - i) + 64
      acc[r] = (t >= 0 && t <= 128) ? ocL[t * 8 + h] : 0.0f;
    }
#pragma unroll
    for (int kk = 0; kk < 16; ++kk) {
      v2f bb = frag_rows(kbuf, ycol0, kk * 4, LDW, lane);  // B[k][n] = k'[y][k]
      acc = wmma4(qf[kk], bb, acc);
    }
#pragma unroll
    for (int r = 0; r < 8; ++r) {
      int m = r + 8 * hi;
      int t = 16 * ty + nl - m;
      int jglob = base_j + ycol0 + nl;
      bool ok = (t >= 0 && t <= 128) && (jglob >= 0 && jglob < kS);
      float val = ok ? tanhf(acc[r]) : 0.0f;
      myslab[m * SLAB_LD + 16 * ty + nl] = val;
      if (t >= 0 && t <= 128) {
        int ig = c * kW + x0w + m;
        attn_out[(((size_t)(b * kS + ig)) * 129 + t) * 8 + h] = val;
      }
    }
  }

  wait_async0();        // v window resident
  __syncthreads();      // slabs visible

  // ================= phase 3: out += slab(16x144) @ v(144x64) ================
  for (int nt = 0; nt < 4; ++nt) {
    const int n0 = nt * 16;
    v8f acc;
#pragma unroll
    for (int r = 0; r < 8; ++r) acc[r] = 0.0f;
#pragma unroll
    for (int k0 = 0; k0 < 144; k0 += 4) {
      v2f a = frag_rows(myslab, 0, k0, SLAB_LD, lane);
      v2f bb = frag_cols(vbuf, x0w + k0, n0, LDW, lane);
      acc = wmma4(a, bb, acc);
    }
#pragma unroll
    for (int r = 0; r < 8; ++r) {
      int ig = c * kW + x0w + r + 8 * hi;
      atomicAdd(&out[(size_t)(b * kS + ig) * kO + n0 + nl], acc[r]);
    }
  }
}

extern "C" void kernel_launch(void* const* d_in, const int* in_sizes, int n_in,
                              void* d_out, int out_size, void* d_ws, size_t ws_size,
                              hipStream_t stream) {
  const float* q = (const float*)d_in[0];
  const float* k = (const float*)d_in[1];
  const float* v = (const float*)d_in[2];
  const float* Wq = (const float*)d_in[3];
  const float* bq = (const float*)d_in[4];
  const float* Wk = (const float*)d_in[5];
  const float* bk = (const float*)d_in[6];
  const float* oc = (const float*)d_in[7];

  float* attn_out = (float*)d_out;                         // (B,S,129,H)
  float* out = attn_out + (size_t)kB * kS * 129 * kH;      // (B,S,O)
  const int n_out = kB * kS * kO;

  zero_kernel<<<(n_out + 255) / 256, 256, 0, stream>>>(out, n_out);

  const size_t smem_bytes = (size_t)SMEM_FLOATS * sizeof(float);
  hipFuncSetAttribute(reinterpret_cast<const void*>(banded_attn_kernel),
                      hipFuncAttributeMaxDynamicSharedMemorySize, (int)smem_bytes);

  banded_attn_kernel<<<kB * kC * kH, 256, smem_bytes, stream>>>(
      q, k, v, Wq, bq, Wk, bk, oc, attn_out, out);
}